// bidirectionalRNN_84404697301453
// MI455X (gfx1250) — compile-verified
//
#include <hip/hip_runtime.h>
#include <cstddef>
#include <cstdint>

// ---------------------------------------------------------------------------
// Bidirectional 2-layer LSTM on MI455X (gfx1250), fp32 via V_WMMA_F32_16X16X4_F32
// + async direct-to-LDS staging (ASYNCcnt) + hardware V_TANH_F32 activations.
// S=256, B=32, I=H=1024, 4H=4096, L=2, 2 directions.
// ---------------------------------------------------------------------------

typedef float v2f __attribute__((ext_vector_type(2)));
typedef float v8f __attribute__((ext_vector_type(8)));

#define SLEN   256
#define BATCH  32
#define HID    1024
#define G4     4096            // 4*H
#define MROWS  8192            // S*B

__device__ __forceinline__ v8f wmma_f32(v2f a, v2f b, v8f c) {
  // D = A(16x4) * B(4x16) + C(16x16), fp32 XDL path on CDNA5
  return __builtin_amdgcn_wmma_f32_16x16x4_f32(false, a, false, b, (short)0, c,
                                               false, false);
}

// Async global->LDS copy, 16B per lane, tracked by ASYNCcnt (CDNA5 TDM-lite path).
__device__ __forceinline__ void async_ld_b128(unsigned lds_off, const float* g) {
  asm volatile("global_load_async_to_lds_b128 %0, %1, off"
               :: "v"(lds_off), "v"(g)
               : "memory");
}
#define ASYNC_WAIT0() asm volatile("s_wait_asynccnt 0x0" ::: "memory")

__device__ __forceinline__ unsigned lds_off(const float* p) {
  // generic pointer to LDS: low 32 bits are the wave-relative LDS byte offset
  return (unsigned)(uintptr_t)p;
}

// Hardware transcendental tanh (TRANS32). s_delay_alu covers the TRANS RAW
// hazard (1 independent op required before consuming a TRANS result).
__device__ __forceinline__ float htanh(float x) {
  float r;
  asm volatile("v_tanh_f32 %0, %1\n\t"
               "s_delay_alu instid0(TRANS32_DEP_1)"
               : "=v"(r) : "v"(x));
  return r;
}
__device__ __forceinline__ float hsig(float x) {
  return __builtin_fmaf(0.5f, htanh(0.5f * x), 0.5f);
}

// ---------------------------------------------------------------------------
// Kernel 1: repack W_hh[4H][H] into WMMA B-fragment order so the step kernel
// reads one coalesced 256B block per (ntile,kstep) fragment.
// pack[(ntile*256 + kstep)*64 + lane*2 + {0,1}]
//   lane<16 :  w[(ntile*16 + lane)  *H + kstep*4 + 0/1]
//   lane>=16:  w[(ntile*16 + lane-16)*H + kstep*4 + 2/3]
// ---------------------------------------------------------------------------
__global__ __launch_bounds__(256) void repack_whh(const float* __restrict__ whF,
                                                  const float* __restrict__ whB,
                                                  float* __restrict__ pack) {
  const int d = blockIdx.y;
  const float* w = d ? whB : whF;
  float* p = pack + (size_t)d * G4 * HID;
  const int slot = blockIdx.x * 256 + threadIdx.x;   // 0 .. 2M-1
  const int lane = slot & 31;
  const int ks   = (slot >> 5) & 255;                // kstep 0..255
  const int nt   = slot >> 13;                       // ntile 0..255
  const int n = nt * 16 + (lane & 15);
  const int k = ks * 4 + ((lane >> 4) << 1);
  p[(size_t)slot * 2 + 0] = w[(size_t)n * HID + k + 0];
  p[(size_t)slot * 2 + 1] = w[(size_t)n * HID + k + 1];
}

// ---------------------------------------------------------------------------
// Kernel 2: xg[d] = X[d] @ W_ih^T   (M=8192, N=4096, K=1024)
// Block: 256 thr (8 waves). WG tile 128x64, wave tile 32x32 (2x2 WMMA tiles).
// Double-buffered LDS, filled by async global->LDS loads overlapped with WMMA.
// ---------------------------------------------------------------------------
#define GPAD 20
__global__ __launch_bounds__(256) void gemm_xg(const float* __restrict__ X0,
                                               const float* __restrict__ X1,
                                               const float* __restrict__ W0,
                                               const float* __restrict__ W1,
                                               float* __restrict__ xg) {
  __shared__ float ldsA[2][128 * GPAD];
  __shared__ float ldsB[2][64 * GPAD];

  const int d = blockIdx.z;
  const float* X = d ? X1 : X0;
  const float* W = d ? W1 : W0;
  float* out = xg + (size_t)d * MROWS * G4;

  const int mBase = blockIdx.x * 128;
  const int nBase = blockIdx.y * 64;
  const int tid = threadIdx.x;
  const int lane = tid & 31;
  const int wv = tid >> 5;
  const int m0 = (wv >> 1) * 32;   // wave M offset in WG tile
  const int n0 = (wv & 1) * 32;    // wave N offset in WG tile

  // staging assignment: 2 b128 for A (128x16), 1 b128 for B (64x16) per thread
  const int ra = tid >> 1, koa = (tid & 1) * 8;
  const int rb = tid >> 2, kob = (tid & 3) * 4;
  const float* gA = X + (size_t)(mBase + ra) * HID + koa;
  const float* gB = W + (size_t)(nBase + rb) * HID + kob;
  const unsigned lA0 = lds_off(&ldsA[0][ra * GPAD + koa]);
  const unsigned lB0 = lds_off(&ldsB[0][rb * GPAD + kob]);
  const unsigned lA1 = lds_off(&ldsA[1][ra * GPAD + koa]);
  const unsigned lB1 = lds_off(&ldsB[1][rb * GPAD + kob]);

  v8f acc[2][2] = {};

  // prologue: chunk 0 into buffer 0
  async_ld_b128(lA0, gA);
  async_ld_b128(lA0 + 16, gA + 4);
  async_ld_b128(lB0, gB);

  const int klo = (lane >> 4) << 1;
  const int l15 = lane & 15;

  for (int ci = 0; ci < 64; ++ci) {
    const int buf = ci & 1;
    ASYNC_WAIT0();         // chunk ci resident in LDS (this wave's async ops)
    __syncthreads();       // ... and everyone else's; prior compute done too
    if (ci < 63) {         // launch chunk ci+1 into the other buffer
      const int kc = (ci + 1) * 16;
      if (buf == 0) {
        async_ld_b128(lA1, gA + kc);
        async_ld_b128(lA1 + 16, gA + kc + 4);
        async_ld_b128(lB1, gB + kc);
      } else {
        async_ld_b128(lA0, gA + kc);
        async_ld_b128(lA0 + 16, gA + kc + 4);
        async_ld_b128(lB0, gB + kc);
      }
    }
    const float* bufA = ldsA[buf];
    const float* bufB = ldsB[buf];
#pragma unroll
    for (int kk = 0; kk < 16; kk += 4) {
      v2f a0 = *(const v2f*)&bufA[(m0 + l15) * GPAD + kk + klo];
      v2f a1 = *(const v2f*)&bufA[(m0 + 16 + l15) * GPAD + kk + klo];
      v2f b0 = *(const v2f*)&bufB[(n0 + l15) * GPAD + kk + klo];
      v2f b1 = *(const v2f*)&bufB[(n0 + 16 + l15) * GPAD + kk + klo];
      acc[0][0] = wmma_f32(a0, b0, acc[0][0]);
      acc[0][1] = wmma_f32(a0, b1, acc[0][1]);
      acc[1][0] = wmma_f32(a1, b0, acc[1][0]);
      acc[1][1] = wmma_f32(a1, b1, acc[1][1]);
    }
    __syncthreads();       // all waves done reading buf before it's refilled
  }

  // epilogue: C/D layout -> VGPR r holds rows (base+r) lanes0-15, (base+r+8) lanes16-31
#pragma unroll
  for (int mt = 0; mt < 2; ++mt)
#pragma unroll
    for (int nt = 0; nt < 2; ++nt) {
      const int col = nBase + n0 + nt * 16 + (lane & 15);
      const int rowb = mBase + m0 + mt * 16 + ((lane >> 4) << 3);
#pragma unroll
      for (int r = 0; r < 8; ++r)
        out[(size_t)(rowb + r) * G4 + col] = acc[mt][nt][r];
    }
}

// ---------------------------------------------------------------------------
// Kernel 3: one LSTM timestep for both directions.
// grid = (16 hidden-chunks of 64, 2 directions); block = 256 thr (8 waves).
// Each WG computes gates[4][32][64] = h_prev @ W_hh^T (WMMA, packed weights),
// adds xg, does the pointwise LSTM cell, writes h (and c state).
// wave w: gate = w&3, mtile = w>>2 (batch 0-15 / 16-31), 4 N-tiles each.
// h_prev staged via double-buffered async global->LDS loads.
// ---------------------------------------------------------------------------
#define HPAD 68
__global__ __launch_bounds__(256) void lstm_step(const float* __restrict__ xg,
                                                 const float* __restrict__ pack,
                                                 float* __restrict__ hs,
                                                 float* __restrict__ out,
                                                 float* __restrict__ cst,
                                                 float* __restrict__ hAll,
                                                 float* __restrict__ cAll,
                                                 int l, int t) {
  __shared__ float ldsH[2][BATCH * HPAD];  // h_prev K-chunk [32][64] (+pad), x2
  __shared__ float ldsG[4 * BATCH * 64];   // gate pre-activations [4][32][64]

  const int d = blockIdx.y;                // 0 = fwd, 1 = bwd
  const int j0 = blockIdx.x * 64;          // hidden-column chunk
  const int tid = threadIdx.x;
  const int lane = tid & 31;
  const int wv = tid >> 5;
  const int gate = wv & 3;
  const int mtile = wv >> 2;

  const float* xgd = xg + (size_t)d * MROWS * G4;
  const float* packd = pack + (size_t)d * G4 * HID;
  float* cstate = cst + (size_t)d * BATCH * HID;

  float* hb;                                // hidden history base (read+write)
  int hstr;
  if (l == 0) { hb = hs + (size_t)d * MROWS * HID; hstr = HID; }
  else        { hb = out + (size_t)d * HID;        hstr = 2 * HID; }

  int s, sprev;
  if (d == 0) { s = t;        sprev = t - 1;   }
  else        { s = 255 - t;  sprev = 256 - t; }

  v8f acc[4] = {};

  if (t > 0) {
    const float* hprev = hb + (size_t)sprev * BATCH * hstr;
    const int l15 = lane & 15;
    const int klo = (lane >> 4) << 1;

    // staging: 2 b128 per thread per 32x64 chunk
    const int rh = tid >> 3;               // 0..31 (batch row)
    const int koh = (tid & 7) * 8;         // 0..56
    const float* gH = hprev + (size_t)rh * hstr + koh;
    const unsigned lH0 = lds_off(&ldsH[0][rh * HPAD + koh]);
    const unsigned lH1 = lds_off(&ldsH[1][rh * HPAD + koh]);

    async_ld_b128(lH0, gH);
    async_ld_b128(lH0 + 16, gH + 4);

    for (int ci = 0; ci < 16; ++ci) {      // K chunks of 64
      const int buf = ci & 1;
      ASYNC_WAIT0();
      __syncthreads();
      if (ci < 15) {
        const int kc = (ci + 1) * 64;
        if (buf == 0) {
          async_ld_b128(lH1, gH + kc);
          async_ld_b128(lH1 + 16, gH + kc + 4);
        } else {
          async_ld_b128(lH0, gH + kc);
          async_ld_b128(lH0 + 16, gH + kc + 4);
        }
      }

      const float* aRow = &ldsH[buf][(mtile * 16 + l15) * HPAD + klo];
      // packed-weight base for this wave/chunk; ntile stride = 256*64 floats
      const float* pb = packd +
          ((size_t)(gate * 64 + (j0 >> 4)) * 256 + ci * 16) * 64 + lane * 2;
#pragma unroll
      for (int kk = 0; kk < 16; ++kk) {
        v2f a = *(const v2f*)(aRow + kk * 4);
        const float* pk = pb + kk * 64;
#pragma unroll
        for (int nt = 0; nt < 4; ++nt) {
          v2f b = *(const v2f*)(pk + (size_t)nt * 16384);
          acc[nt] = wmma_f32(a, b, acc[nt]);
        }
      }
      __syncthreads();
    }
  }

  // scatter accumulators to LDS gate buffer
#pragma unroll
  for (int nt = 0; nt < 4; ++nt) {
    const int col = nt * 16 + (lane & 15);
#pragma unroll
    for (int r = 0; r < 8; ++r) {
      const int row = mtile * 16 + r + ((lane >> 4) << 3);
      ldsG[gate * (BATCH * 64) + row * 64 + col] = acc[nt][r];
    }
  }
  __syncthreads();

  // pointwise LSTM cell: 32x64 elements, 8 per thread; V_TANH_F32 activations
#pragma unroll
  for (int e = 0; e < 8; ++e) {
    const int idx = e * 256 + tid;       // 0..2047
    const int b = idx >> 6;
    const int jl = idx & 63;
    const int j = j0 + jl;
    const size_t xrow = ((size_t)s * BATCH + b) * G4;

    float gi = ldsG[0 * 2048 + b * 64 + jl] + xgd[xrow + 0 * HID + j];
    float gf = ldsG[1 * 2048 + b * 64 + jl] + xgd[xrow + 1 * HID + j];
    float gg = ldsG[2 * 2048 + b * 64 + jl] + xgd[xrow + 2 * HID + j];
    float go = ldsG[3 * 2048 + b * 64 + jl] + xgd[xrow + 3 * HID + j];

    const float iv = hsig(gi);
    const float fv = hsig(gf);
    const float gv = htanh(gg);
    const float ov = hsig(go);

    const float cold = (t == 0) ? 0.0f : cstate[b * HID + j];
    const float cn = fv * cold + iv * gv;
    cstate[b * HID + j] = cn;
    const float hn = ov * htanh(cn);

    hb[((size_t)s * BATCH + b) * hstr + j] = hn;

    if (s == SLEN - 1) {  // torch stackedRNN "last" = state at original index S-1
      hAll[((size_t)l * BATCH + b) * (2 * HID) + d * HID + j] = hn;
      cAll[((size_t)l * BATCH + b) * (2 * HID) + d * HID + j] = cn;
    }
  }
}

// ---------------------------------------------------------------------------
// Host launcher
// ---------------------------------------------------------------------------
extern "C" void kernel_launch(void* const* d_in, const int* in_sizes, int n_in,
                              void* d_out, int out_size, void* d_ws, size_t ws_size,
                              hipStream_t stream) {
  (void)in_sizes; (void)n_in; (void)out_size; (void)ws_size;

  const float* x    = (const float*)d_in[0];
  const float* wihF = (const float*)d_in[1];
  const float* whhF = (const float*)d_in[2];
  const float* wihB = (const float*)d_in[3];
  const float* whhB = (const float*)d_in[4];

  float* out  = (float*)d_out;
  float* hAll = out + (size_t)SLEN * BATCH * 2 * HID;      // [2][32][2048]
  float* cAll = hAll + (size_t)2 * BATCH * 2 * HID;

  // workspace carve-up (floats)
  float* ws   = (float*)d_ws;
  float* xg   = ws;                                        // 2 * 8192 * 4096
  float* hs   = xg + (size_t)2 * MROWS * G4;               // 2 * 8192 * 1024
  float* cst  = hs + (size_t)2 * MROWS * HID;              // 2 * 32 * 1024
  float* pack = cst + (size_t)2 * BATCH * HID;             // 2 * 4096 * 1024

  for (int l = 0; l < 2; ++l) {
    const size_t woff = (size_t)l * G4 * HID;              // layer offset in weights

    repack_whh<<<dim3(8192, 2), 256, 0, stream>>>(whhF + woff, whhB + woff, pack);

    const float* X0 = (l == 0) ? x : hs;
    const float* X1 = (l == 0) ? x : hs + (size_t)MROWS * HID;
    gemm_xg<<<dim3(64, 64, 2), 256, 0, stream>>>(X0, X1, wihF + woff, wihB + woff, xg);

    for (int t = 0; t < SLEN; ++t)
      lstm_step<<<dim3(16, 2), 256, 0, stream>>>(xg, pack, hs, out, cst,
                                                 hAll, cAll, l, t);
  }
}